// TemporalConsistencyLoss_45861660787097
// MI455X (gfx1250) — compile-verified
//
#include <hip/hip_runtime.h>

#define B_    8
#define S_    16
#define T_    65536
#define W_    5
#define OWN   252                      // columns owned per block (256 computed, 4 halo)
#define BLK   256
#define NBLKX ((T_ + OWN - 1) / OWN)   // 261
#define NPART (NBLKX * B_)             // 2088 partial pairs

typedef __attribute__((ext_vector_type(2))) float v2f;
typedef __attribute__((ext_vector_type(8))) float v8f;

__global__ __launch_bounds__(BLK) void tcl_main(const float* __restrict__ X,
                                                float* __restrict__ pkl,
                                                float* __restrict__ pvar) {
    const int lt = threadIdx.x;           // local column
    const int b  = blockIdx.y;            // batch
    const int t0 = blockIdx.x * OWN;
    const int t  = t0 + lt;               // global column (time)

    __shared__ float sx [S_][BLK];        // raw logits tile (async-to-LDS target)
    __shared__ float sp [S_][BLK];        // softmax probabilities tile
    __shared__ float slp[S_][BLK];        // log-probability tile
    __shared__ float rkl[BLK];            // per-thread KL partials
    __shared__ float rvar[BLK];           // per-thread window-variance partials

    const bool valid = (t < T_);
    const int  tc    = valid ? t : (T_ - 1);        // clamp halo addresses in-bounds
    const unsigned goff = (unsigned)tc * 4u;        // per-lane byte offset within a row

    // ---- CDNA5 async-to-LDS staging: one coalesced 128B transfer per s-row per
    // wave, tracked on ASYNCcnt. Each lane deposits into its own LDS slot, so
    // only s_wait_asynccnt is needed before this lane consumes its column.
    #pragma unroll
    for (int s = 0; s < S_; ++s) {
        const float* rowbase = X + ((size_t)b * S_ + s) * (size_t)T_;
        unsigned ldsoff = (unsigned)(uintptr_t)&sx[s][lt];   // low 32 bits = LDS offset
        asm volatile("global_load_async_to_lds_b32 %0, %1, %2"
                     :: "v"(ldsoff), "v"(goff), "s"(rowbase)
                     : "memory");
    }
    asm volatile("s_wait_asynccnt 0" ::: "memory");

    float x[S_];
    #pragma unroll
    for (int s = 0; s < S_; ++s) x[s] = sx[s][lt];

    // Softmax over s (max-subtracted); publish p and log p to LDS.
    float m = x[0];
    #pragma unroll
    for (int s = 1; s < S_; ++s) m = fmaxf(m, x[s]);
    float p[S_];
    float E = 0.0f;
    #pragma unroll
    for (int s = 0; s < S_; ++s) { p[s] = __expf(x[s] - m); E += p[s]; }
    const float invE = 1.0f / E;
    const float logE = __logf(E);
    #pragma unroll
    for (int s = 0; s < S_; ++s) {
        p[s] *= invE;
        sp [s][lt] = p[s];
        slp[s][lt] = (x[s] - m) - logE;   // exact log p, no transcendental
    }
    __syncthreads();

    float kl_acc  = 0.0f;
    float var_acc = 0.0f;
    const bool owned = (lt < OWN) && valid;

    // KL(p_t || p_{t+1}) = sum_s p*(log p - log q); log q read from neighbor's
    // published tile (EPS inside the reference's log shifts this by ~1e-7 rel).
    if (owned && t <= T_ - 2) {
        #pragma unroll
        for (int s = 0; s < S_; ++s) {
            float lq = slp[s][lt + 1];
            float lp = slp[s][lt];
            kl_acc += p[s] * (lp - lq);
        }
    }

    // Sliding-window variance: starts at even t, window [t, t+4], var = SS/(W-1).
    if (owned && ((t & 1) == 0) && t <= T_ - W_) {
        #pragma unroll
        for (int s = 0; s < S_; ++s) {
            float v0 = p[s];
            float v1 = sp[s][lt + 1];
            float v2 = sp[s][lt + 2];
            float v3 = sp[s][lt + 3];
            float v4 = sp[s][lt + 4];
            float mean = (v0 + v1 + v2 + v3 + v4) * 0.2f;
            float d0 = v0 - mean, d1 = v1 - mean, d2 = v2 - mean,
                  d3 = v3 - mean, d4 = v4 - mean;
            var_acc += d0*d0 + d1*d1 + d2*d2 + d3*d3 + d4*d4;  // SS; /4 at the end
        }
    }

    rkl[lt]  = kl_acc;
    rvar[lt] = var_acc;
    __syncthreads();

    // Block reduction on the matrix pipe: V_WMMA_F32_16X16X4_F32 (full fp32).
    // A (16x4) is a 0/1 selector; B VGPR0 carries 32 kl partials, VGPR1 carries
    // 32 var partials. D[0][n] accumulates kl pair-sums, D[1][n] var pair-sums;
    // 8 chained WMMAs with C-accumulate consume all 256 partials.
    if (lt < 32) {                       // wave 0, EXEC all ones inside the wave
        const int lane = lt;
        v2f a;
        a.x = ((lane & 15) == 0) ? 1.0f : 0.0f;   // selector row M=0
        a.y = ((lane & 15) == 1) ? 1.0f : 0.0f;   // selector row M=1
        v8f acc = {};
        #pragma unroll
        for (int i = 0; i < 8; ++i) {
            v2f bm;
            bm.x = rkl[i * 32 + lane];
            bm.y = rvar[i * 32 + lane];
            acc = __builtin_amdgcn_wmma_f32_16x16x4_f32(
                      false, a, false, bm, (short)0, acc, false, false);
        }
        float ksum = acc[0];             // D row M=0, lanes 0..15
        float vsum = acc[1];             // D row M=1, lanes 0..15
        #pragma unroll
        for (int off = 8; off > 0; off >>= 1) {
            ksum += __shfl_down(ksum, off, 32);
            vsum += __shfl_down(vsum, off, 32);
        }
        if (lane == 0) {
            const int pidx = b * NBLKX + blockIdx.x;
            pkl[pidx]  = ksum;
            pvar[pidx] = vsum;
        }
    }
}

__global__ __launch_bounds__(BLK) void tcl_final(const float* __restrict__ pkl,
                                                 const float* __restrict__ pvar,
                                                 float* __restrict__ out, int n) {
    __shared__ float rk[BLK];
    __shared__ float rv[BLK];
    float k = 0.0f, v = 0.0f;
    for (int i = threadIdx.x; i < n; i += BLK) { k += pkl[i]; v += pvar[i]; }
    rk[threadIdx.x] = k;
    rv[threadIdx.x] = v;
    __syncthreads();
    for (int off = BLK / 2; off > 0; off >>= 1) {
        if ((int)threadIdx.x < off) {
            rk[threadIdx.x] += rk[threadIdx.x + off];
            rv[threadIdx.x] += rv[threadIdx.x + off];
        }
        __syncthreads();
    }
    if (threadIdx.x == 0) {
        const float SW = 0.1f;
        // smoothness = SW * mean_b( sum_{s,t} kl / (T-1) )
        float smooth = SW * rk[0] / ((float)(T_ - 1) * (float)B_);
        // window_consistency = (sum SS / 4) / (B*S) / (T-W+1)
        float wincon = (rv[0] * 0.25f) /
                       ((float)B_ * (float)S_ * (float)(T_ - W_ + 1));
        out[0] = smooth + SW * wincon;
    }
}

extern "C" void kernel_launch(void* const* d_in, const int* in_sizes, int n_in,
                              void* d_out, int out_size, void* d_ws, size_t ws_size,
                              hipStream_t stream) {
    const float* X = (const float*)d_in[0];   // speaker_predictions (B,S,T)
    // d_in[1] (temporal_features) is unused by the reference.
    float* out  = (float*)d_out;
    float* ws   = (float*)d_ws;
    float* pkl  = ws;                 // NPART floats
    float* pvar = ws + NPART;         // NPART floats

    dim3 grid(NBLKX, B_);
    tcl_main<<<grid, BLK, 0, stream>>>(X, pkl, pvar);
    tcl_final<<<1, BLK, 0, stream>>>(pkl, pvar, out, NPART);
}